// HeterogeneousGNN_30975304139126
// MI455X (gfx1250) — compile-verified
//
#include <hip/hip_runtime.h>
#include <hip/hip_bf16.h>

// ---------------------------------------------------------------------------
// HeteroTemporalEncoder + 2-layer HeteroGraphSAGE forward on gfx1250 (MI455X).
// Dense 128x128 GEMMs run on V_WMMA_F32_16X16X4_F32 (fp32-exact).
// Sparse mean-aggregation uses wave-per-edge float4 gathers + f32 atomics
// (node features fit in the 192MB L2, so gathers are L2-resident).
// ---------------------------------------------------------------------------

#define CDIM 128
#define HALFC 64
#define LDSP 132          // padded LDS row stride (floats) to avoid bank conflicts
#define N_USER 100000
#define N_ITEM 200000
#define NEDGE 1500000
#define LAYERS 2

static_assert(N_USER % 16 == 0, "user tile");
static_assert(N_ITEM % 16 == 0, "item tile");

typedef float v2f __attribute__((ext_vector_type(2)));
typedef float v8f __attribute__((ext_vector_type(8)));

// -------------------------------- utility ----------------------------------

__global__ __launch_bounds__(256)
void zero_kernel(float* __restrict__ p, long n) {
  long i = (long)blockIdx.x * blockDim.x + threadIdx.x;
  if (i < n) p[i] = 0.0f;
}

__global__ __launch_bounds__(256)
void count_kernel(const int* __restrict__ dst, float* __restrict__ cnt, int nedges) {
  int e = blockIdx.x * blockDim.x + threadIdx.x;
  if (e < nedges) atomicAdd(&cnt[dst[e]], 1.0f);
}

__global__ __launch_bounds__(256)
void invert_kernel(float* __restrict__ c, int n) {
  int i = blockIdx.x * blockDim.x + threadIdx.x;
  if (i < n) c[i] = 1.0f / fmaxf(c[i], 1.0f);
}

// ---------------------- edge gather + segment scatter-add ------------------
// one wave32 per edge: each lane moves one float4 (512B coalesced per edge)

__global__ __launch_bounds__(256)
void scatter_add_kernel(const float* __restrict__ xsrc,
                        const int* __restrict__ src,
                        const int* __restrict__ dst,
                        float* __restrict__ msum,
                        int nedges) {
  long gid = (long)blockIdx.x * blockDim.x + threadIdx.x;
  int edge = (int)(gid >> 5);
  int lane = (int)(gid & 31);
  if (edge >= nedges) return;
  int s = src[edge];
  int d = dst[edge];
  float4 v = ((const float4*)(xsrc + (long)s * CDIM))[lane];
  float* md = msum + (long)d * CDIM + lane * 4;
  atomicAdd(md + 0, v.x);
  atomicAdd(md + 1, v.y);
  atomicAdd(md + 2, v.z);
  atomicAdd(md + 3, v.w);
}

// ------------------- temporal encoder: x += pe(rel) @ W^T + b --------------
// block = 256 threads (8 waves) = 16 nodes; each wave computes a 16x16 tile
// of pe @ W^T via 32 chained v_wmma_f32_16x16x4_f32.

__global__ __launch_bounds__(256)
void temporal_encode_kernel(const float* __restrict__ x_in,
                            const int* __restrict__ seed_time,
                            const int* __restrict__ node_time,
                            const int* __restrict__ node_batch,
                            const float* __restrict__ W,     // [C,C] row-major
                            const float* __restrict__ bias,  // [C]
                            float* __restrict__ x_out) {
  __shared__ float pe[16][LDSP];
  __shared__ float rel[16];
  const int tid = threadIdx.x;
  const int node0 = blockIdx.x * 16;

  if (tid < 16) {
    int n = node0 + tid;
    rel[tid] = (float)(seed_time[node_batch[n]] - node_time[n]) * (1.0f / 86400.0f);
  }
  __syncthreads();

  // interleaved sin/cos positional encoding: pe[n][2h]=sin, pe[n][2h+1]=cos
  for (int p = tid; p < 16 * HALFC; p += 256) {
    int nr = p >> 6;
    int h = p & 63;
    // div[h] = 10000^(-h/64) = exp(-h*ln(10000)/64)
    float ang = rel[nr] * __expf(-0.14391156595f * (float)h);
    float s, c;
    __sincosf(ang, &s, &c);
    pe[nr][2 * h] = s;
    pe[nr][2 * h + 1] = c;
  }
  __syncthreads();

  const int wave = tid >> 5;
  const int lane = tid & 31;
  const int half = lane >> 4;   // 0: K lanes {0,1}, 1: K lanes {2,3}
  const int l16 = lane & 15;
  const int col = wave * 16 + l16;     // output feature (N index)
  const int ksel = half * 2;
  const float* Wrow = W + (long)col * CDIM;  // (pe @ W^T)[:,col] uses W[col,:]

  v8f acc = {};
  for (int kk = 0; kk < CDIM; kk += 4) {
    v2f a, b;
    a[0] = pe[l16][kk + ksel];
    a[1] = pe[l16][kk + ksel + 1];
    b[0] = Wrow[kk + ksel];
    b[1] = Wrow[kk + ksel + 1];
    acc = __builtin_amdgcn_wmma_f32_16x16x4_f32(false, a, false, b,
                                                (short)0, acc, false, false);
  }

  // acc vgpr g -> row M = g + 8*half, column N = col
  float bcol = bias[col];
  for (int g = 0; g < 8; ++g) {
    int n = node0 + g + 8 * half;
    x_out[(long)n * CDIM + col] = acc[g] + x_in[(long)n * CDIM + col] + bcol;
  }
}

// --------- SAGE layer: x = relu(LN(m*inv_deg @ Wl^T + bl + x @ Wr^T)) ------

__global__ __launch_bounds__(256)
void sage_update_kernel(float* __restrict__ x,          // [n,C] in/out
                        const float* __restrict__ m,    // [n,C] summed messages
                        const float* __restrict__ invc, // [n] 1/max(deg,1)
                        const float* __restrict__ Wl,   // [C,C]
                        const float* __restrict__ Wr,   // [C,C]
                        const float* __restrict__ bl,   // [C]
                        const float* __restrict__ lng,  // [C]
                        const float* __restrict__ lnb)  // [C]
{
  __shared__ float sm[16][LDSP];
  __shared__ float sx[16][LDSP];
  __shared__ float so[16][LDSP];
  const int tid = threadIdx.x;
  const int node0 = blockIdx.x * 16;

  for (int idx = tid; idx < 16 * CDIM; idx += 256) {
    int r = idx >> 7;
    int f = idx & 127;
    int n = node0 + r;
    sm[r][f] = m[(long)n * CDIM + f] * invc[n];
    sx[r][f] = x[(long)n * CDIM + f];
  }
  __syncthreads();

  const int wave = tid >> 5;
  const int lane = tid & 31;
  const int half = lane >> 4;
  const int l16 = lane & 15;
  const int col = wave * 16 + l16;
  const int ksel = half * 2;
  const float* Wlr = Wl + (long)col * CDIM;
  const float* Wrr = Wr + (long)col * CDIM;

  v8f acc = {};
  for (int kk = 0; kk < CDIM; kk += 4) {
    v2f a, b;
    a[0] = sm[l16][kk + ksel];
    a[1] = sm[l16][kk + ksel + 1];
    b[0] = Wlr[kk + ksel];
    b[1] = Wlr[kk + ksel + 1];
    acc = __builtin_amdgcn_wmma_f32_16x16x4_f32(false, a, false, b,
                                                (short)0, acc, false, false);
    a[0] = sx[l16][kk + ksel];
    a[1] = sx[l16][kk + ksel + 1];
    b[0] = Wrr[kk + ksel];
    b[1] = Wrr[kk + ksel + 1];
    acc = __builtin_amdgcn_wmma_f32_16x16x4_f32(false, a, false, b,
                                                (short)0, acc, false, false);
  }

  float bcol = bl[col];
  for (int g = 0; g < 8; ++g) {
    so[g + 8 * half][col] = acc[g] + bcol;
  }
  __syncthreads();

  // LayerNorm + ReLU: each wave handles 2 nodes (16 lanes per node)
  int r = wave * 2 + half;
  float s = 0.0f, s2 = 0.0f;
  for (int i = 0; i < 8; ++i) {
    float v = so[r][l16 + i * 16];
    s += v;
    s2 += v * v;
  }
  for (int off = 1; off < 16; off <<= 1) {   // xor 1,2,4,8 stays within 16-lane group
    s += __shfl_xor(s, off, 32);
    s2 += __shfl_xor(s2, off, 32);
  }
  float mu = s * (1.0f / 128.0f);
  float var = s2 * (1.0f / 128.0f) - mu * mu;
  float rstd = rsqrtf(var + 1e-5f);
  int n = node0 + r;
  for (int i = 0; i < 8; ++i) {
    int f = l16 + i * 16;
    float v = (so[r][f] - mu) * rstd * lng[f] + lnb[f];
    x[(long)n * CDIM + f] = fmaxf(v, 0.0f);
  }
}

// ------------------------------- launcher ----------------------------------

extern "C" void kernel_launch(void* const* d_in, const int* in_sizes, int n_in,
                              void* d_out, int out_size, void* d_ws, size_t ws_size,
                              hipStream_t stream) {
  (void)in_sizes; (void)n_in; (void)out_size; (void)ws_size;

  const float* x_user = (const float*)d_in[0];
  const float* x_item = (const float*)d_in[1];
  const int* seed_time = (const int*)d_in[2];
  const int* time_user = (const int*)d_in[3];
  const int* time_item = (const int*)d_in[4];
  const int* batch_user = (const int*)d_in[5];
  const int* batch_item = (const int*)d_in[6];
  const int* src_ui = (const int*)d_in[7];
  const int* dst_ui = (const int*)d_in[8];
  const int* src_iu = (const int*)d_in[9];
  const int* dst_iu = (const int*)d_in[10];
  const float* te_W_user = (const float*)d_in[11];
  const float* te_b_user = (const float*)d_in[12];
  const float* te_W_item = (const float*)d_in[13];
  const float* te_b_item = (const float*)d_in[14];
  const float* Wl_ui = (const float*)d_in[15];
  const float* bl_ui = (const float*)d_in[16];
  const float* Wr_ui = (const float*)d_in[17];
  const float* Wl_iu = (const float*)d_in[18];
  const float* bl_iu = (const float*)d_in[19];
  const float* Wr_iu = (const float*)d_in[20];
  const float* ln_g_user = (const float*)d_in[21];
  const float* ln_b_user = (const float*)d_in[22];
  const float* ln_g_item = (const float*)d_in[23];
  const float* ln_b_item = (const float*)d_in[24];

  // live node features reside directly in d_out (= concat(x_user, x_item))
  float* xu = (float*)d_out;
  float* xi = xu + (long)N_USER * CDIM;

  // workspace: message accumulators + inverse-degree vectors
  float* m_user = (float*)d_ws;                    // N_USER*C
  float* m_item = m_user + (long)N_USER * CDIM;    // N_ITEM*C
  float* inv_user = m_item + (long)N_ITEM * CDIM;  // N_USER
  float* inv_item = inv_user + N_USER;             // N_ITEM

  const long mu_n = (long)N_USER * CDIM;
  const long mi_n = (long)N_ITEM * CDIM;

  // ---- degrees (edges are static across layers; compute once per call) ----
  zero_kernel<<<(N_USER + 255) / 256, 256, 0, stream>>>(inv_user, N_USER);
  zero_kernel<<<(N_ITEM + 255) / 256, 256, 0, stream>>>(inv_item, N_ITEM);
  count_kernel<<<(NEDGE + 255) / 256, 256, 0, stream>>>(dst_ui, inv_item, NEDGE);
  count_kernel<<<(NEDGE + 255) / 256, 256, 0, stream>>>(dst_iu, inv_user, NEDGE);
  invert_kernel<<<(N_USER + 255) / 256, 256, 0, stream>>>(inv_user, N_USER);
  invert_kernel<<<(N_ITEM + 255) / 256, 256, 0, stream>>>(inv_item, N_ITEM);

  // ---- temporal encoder (WMMA) ----
  temporal_encode_kernel<<<N_USER / 16, 256, 0, stream>>>(
      x_user, seed_time, time_user, batch_user, te_W_user, te_b_user, xu);
  temporal_encode_kernel<<<N_ITEM / 16, 256, 0, stream>>>(
      x_item, seed_time, time_item, batch_item, te_W_item, te_b_item, xi);

  // ---- 2 SAGE layers ----
  const long sgrid = ((long)NEDGE * 32 + 255) / 256;
  for (int l = 0; l < LAYERS; ++l) {
    zero_kernel<<<(mi_n + 255) / 256, 256, 0, stream>>>(m_item, mi_n);
    zero_kernel<<<(mu_n + 255) / 256, 256, 0, stream>>>(m_user, mu_n);
    scatter_add_kernel<<<(int)sgrid, 256, 0, stream>>>(xu, src_ui, dst_ui, m_item, NEDGE);
    scatter_add_kernel<<<(int)sgrid, 256, 0, stream>>>(xi, src_iu, dst_iu, m_user, NEDGE);
    sage_update_kernel<<<N_ITEM / 16, 256, 0, stream>>>(
        xi, m_item, inv_item,
        Wl_ui + (long)l * CDIM * CDIM, Wr_ui + (long)l * CDIM * CDIM,
        bl_ui + (long)l * CDIM, ln_g_item + (long)l * CDIM, ln_b_item + (long)l * CDIM);
    sage_update_kernel<<<N_USER / 16, 256, 0, stream>>>(
        xu, m_user, inv_user,
        Wl_iu + (long)l * CDIM * CDIM, Wr_iu + (long)l * CDIM * CDIM,
        bl_iu + (long)l * CDIM, ln_g_user + (long)l * CDIM, ln_b_user + (long)l * CDIM);
  }
}